// Attention1_85581518340102
// MI455X (gfx1250) — compile-verified
//
#include <hip/hip_runtime.h>
#include <math.h>

#define BD 4
#define DIM 96
#define NH 8
#define CHD 12
#define HT 160
#define WDT 160
#define HW 25600
#define C3 288

typedef float v2f __attribute__((ext_vector_type(2)));
typedef float v8f __attribute__((ext_vector_type(8)));

__device__ __forceinline__ float gelu_f(float v) {
    return 0.5f * v * (1.0f + erff(v * 0.70710678118654752f));
}
__device__ __forceinline__ float sigmoid_f(float v) {
    return 1.0f / (1.0f + expf(-v));
}

__device__ __forceinline__ float block_reduce_sum(float v, float* red) {
    int t = threadIdx.x;
    red[t] = v;
    __syncthreads();
    for (int s = 128; s > 0; s >>= 1) {
        if (t < s) red[t] += red[t + s];
        __syncthreads();
    }
    return red[0];
}

// ---------------- K1: per-channel sums of x (for cm) ----------------
__global__ void k_chansum(const float* __restrict__ x, float* __restrict__ chan_sum) {
    __shared__ float red[256];
    int rowi = blockIdx.x;                      // b*DIM + c
    const float* p = x + (size_t)rowi * HW;
    float s = 0.f;
    for (int i = threadIdx.x; i < HW; i += 256) s += p[i];
    float tot = block_reduce_sum(s, red);
    if (threadIdx.x == 0) chan_sum[rowi] = tot;
}

// ---------------- K2: SE gate, folded weights, saw (analytic), sgm ----------------
__global__ void k_prep(const float* __restrict__ spectral, const float* __restrict__ chan_sum,
                       const float* __restrict__ w_se1, const float* __restrict__ b_se1,
                       const float* __restrict__ w_se2, const float* __restrict__ b_se2,
                       const float* __restrict__ w_qkv,
                       const float* __restrict__ w_saw1, const float* __restrict__ b_saw1,
                       const float* __restrict__ w_saw2, const float* __restrict__ b_saw2,
                       const float* __restrict__ w_sgp, const float* __restrict__ b_sgp,
                       float* __restrict__ W_eff, float* __restrict__ sgm_out) {
    __shared__ float sp[DIM], hid[2 * DIM], sw[DIM], esg[DIM], saw_s[NH];
    __shared__ float cm_s;
    int b = blockIdx.x, t = threadIdx.x;
    if (t < DIM) sp[t] = spectral[b * DIM + t];
    __syncthreads();
    if (t < 2 * DIM) {
        float a = b_se1[t];
        for (int i = 0; i < DIM; i++) a += sp[i] * w_se1[t * DIM + i];
        hid[t] = gelu_f(a);
    }
    __syncthreads();
    if (t < DIM) {
        float a = b_se2[t];
        for (int o = 0; o < 2 * DIM; o++) a += hid[o] * w_se2[t * 2 * DIM + o];
        sw[t] = sigmoid_f(a);
    }
    __syncthreads();
    if (t == 0) {
        float s = 0.f;
        for (int c = 0; c < DIM; c++) s += sw[c] * chan_sum[b * DIM + c];
        cm_s = s / (float)(DIM * HW);
    }
    // fold SE gate into qkv weights: W_eff[oc,ic] = w_qkv[oc,ic]*sw[ic]
    for (int i = t; i < C3 * DIM; i += 256)
        W_eff[(size_t)b * C3 * DIM + i] = w_qkv[i] * sw[i % DIM];
    __syncthreads();
    float cm = cm_s;
    // saw: conv over a constant field -> 9 border regions, exact weighted average
    if (t < NH) {
        float acc = 0.f;
        for (int ry = 0; ry < 3; ry++) {
            int ky0 = (ry == 0) ? 1 : 0, ky1 = (ry == 2) ? 1 : 2;
            float cy = (ry == 1) ? 158.f : 1.f;
            for (int cx = 0; cx < 3; cx++) {
                int kx0 = (cx == 0) ? 1 : 0, kx1 = (cx == 2) ? 1 : 2;
                float cnt = cy * ((cx == 1) ? 158.f : 1.f);
                float z = b_saw2[t];
                for (int oc = 0; oc < 24; oc++) {
                    float S = 0.f;
                    for (int ky = ky0; ky <= ky1; ky++)
                        for (int kx = kx0; kx <= kx1; kx++)
                            S += w_saw1[oc * 9 + ky * 3 + kx];
                    z += w_saw2[t * 24 + oc] * gelu_f(b_saw1[oc] + cm * S);
                }
                acc += cnt * sigmoid_f(z);
            }
        }
        saw_s[t] = acc / (float)HW;
    }
    __syncthreads();
    if (t < DIM) {
        float a = b_sgp[t];
        for (int i = 0; i < DIM; i++) a += sp[i] * w_sgp[t * DIM + i];
        esg[t] = a * saw_s[t / CHD];
    }
    __syncthreads();
    for (int i = t; i < NH * 144; i += 256) {
        int h = i / 144, r = i % 144;
        sgm_out[(size_t)b * 1152 + i] = esg[h * CHD + r / 12] * esg[h * CHD + r % 12];
    }
}

// ---------------- WMMA fp32 GEMM: Y[b] = W[b] (M x 96) @ X[b] (96 x 25600) (+Add) ----------------
__global__ void k_wmma_gemm96(const float* __restrict__ Wm, const float* __restrict__ Xm,
                              const float* __restrict__ Addm, float* __restrict__ Ym,
                              int tilesM, long sW, long sX, long sA, long sY) {
    int wv = threadIdx.x >> 5;
    int lane = threadIdx.x & 31;
    int tile = blockIdx.x * 8 + wv;
    int total = tilesM * (HW / 16);
    if (tile >= total) return;
    int tm = tile % tilesM, tn = tile / tilesM;
    int m0 = tm * 16, n0 = tn * 16;
    int row = lane & 15, hi = lane >> 4;
    const float* Wp = Wm + (size_t)blockIdx.y * sW + (size_t)(m0 + row) * 96;
    const float* Xp = Xm + (size_t)blockIdx.y * sX + n0 + row;
    v8f acc = {0.f, 0.f, 0.f, 0.f, 0.f, 0.f, 0.f, 0.f};
#pragma unroll
    for (int k = 0; k < 96; k += 4) {
        v2f a, bf;
        // A 16x4: VGPR0 = K (lane<16: k+0, lane>=16: k+2), VGPR1 = next K
        a.x = Wp[k + 2 * hi];
        a.y = Wp[k + 2 * hi + 1];
        // B 4x16: rows striped across lanes within a VGPR
        bf.x = Xp[(size_t)(k + 2 * hi) * HW];
        bf.y = Xp[(size_t)(k + 2 * hi + 1) * HW];
        acc = __builtin_amdgcn_wmma_f32_16x16x4_f32(false, a, false, bf,
                                                    (short)0, acc, false, false);
    }
    float* Yp = Ym + (size_t)blockIdx.y * sY + (size_t)m0 * HW + n0 + row;
    if (Addm) {
        const float* Ap = Addm + (size_t)blockIdx.y * sA + (size_t)m0 * HW + n0 + row;
#pragma unroll
        for (int v = 0; v < 8; v++)
            Yp[(size_t)(v + 8 * hi) * HW] = acc[v] + Ap[(size_t)(v + 8 * hi) * HW];
    } else {
#pragma unroll
        for (int v = 0; v < 8; v++)
            Yp[(size_t)(v + 8 * hi) * HW] = acc[v];
    }
}

// ---------------- K4: 3x3 depthwise conv (288 channels) ----------------
__global__ void k_dwconv(const float* __restrict__ qkv_pre, const float* __restrict__ w_dw,
                         float* __restrict__ qkv) {
    int bc = blockIdx.y;            // b*C3 + ch
    int ch = bc % C3;
    const float* wp = w_dw + ch * 9;  // uniform -> scalar loads
    int pix = blockIdx.x * 256 + threadIdx.x;
    int y = pix / WDT, xp = pix % WDT;
    const float* in = qkv_pre + (size_t)bc * HW;
    float acc = 0.f;
#pragma unroll
    for (int ky = 0; ky < 3; ky++) {
        int yy = y + ky - 1;
        if (yy < 0 || yy >= HT) continue;
#pragma unroll
        for (int kx = 0; kx < 3; kx++) {
            int xx = xp + kx - 1;
            if (xx < 0 || xx >= WDT) continue;
            acc += wp[ky * 3 + kx] * in[yy * WDT + xx];
        }
    }
    qkv[(size_t)bc * HW + pix] = acc;
}

// ---------------- K5: fc branch (reinterpret-view 9x24 matmul per 288-elem group) ----------------
__global__ void k_fc(const float* __restrict__ qkv, const float* __restrict__ w_fc,
                     const float* __restrict__ b_fc, float* __restrict__ f_conv) {
    int b = blockIdx.y;
    int t = threadIdx.x;            // 192 threads: 16 groups x 12 d
    int d = t % 12;
    int n = blockIdx.x * 16 + t / 12;
    const float* raw = qkv + (size_t)b * C3 * HW + (size_t)n * C3 + d;
    float m[24];
#pragma unroll
    for (int c = 0; c < 24; c++) m[c] = raw[c * 12];
    float* fo = f_conv + (size_t)b * 108 * HW;
#pragma unroll
    for (int o = 0; o < 9; o++) {
        float acc = b_fc[o];
#pragma unroll
        for (int c = 0; c < 24; c++) acc += w_fc[o * 24 + c] * m[c];
        fo[(size_t)(d * 9 + o) * HW + n] = acc;
    }
}

// ---------------- K6: grouped 3x3 conv (12 groups, 9 -> 8 channels) ----------------
__global__ void k_depconv(const float* __restrict__ f_conv, const float* __restrict__ w_dep,
                          const float* __restrict__ b_dep, float* __restrict__ out_conv) {
    int bo = blockIdx.y;            // b*DIM + oc
    int b = bo / DIM, oc = bo % DIM;
    int g = oc / 8;
    int pix = blockIdx.x * 256 + threadIdx.x;
    int y = pix / WDT, xp = pix % WDT;
    const float* fin = f_conv + (size_t)b * 108 * HW + (size_t)(g * 9) * HW;
    const float* wp = w_dep + oc * 81;   // uniform per block
    float acc = b_dep[oc];
    for (int ic = 0; ic < 9; ic++) {
        const float* f2 = fin + (size_t)ic * HW;
#pragma unroll
        for (int ky = 0; ky < 3; ky++) {
            int yy = y + ky - 1;
            if (yy < 0 || yy >= HT) continue;
#pragma unroll
            for (int kx = 0; kx < 3; kx++) {
                int xx = xp + kx - 1;
                if (xx < 0 || xx >= WDT) continue;
                acc += wp[ic * 9 + ky * 3 + kx] * f2[yy * WDT + xx];
            }
        }
    }
    out_conv[(size_t)bo * HW + pix] = acc;
}

// ---------------- K7: gram matrices + row sq-norms (reductions over 25600) ----------------
__global__ void k_gram(const float* __restrict__ qkv, float* __restrict__ G) {
    __shared__ float red[256];
    int p = blockIdx.x;             // 0..143 q.k ; 144..155 q.q ; 156..167 k.k
    int bh = blockIdx.y;
    int b = bh / NH, h = bh % NH;
    const float* base = qkv + (size_t)b * C3 * HW;
    const float *pa, *pb;
    if (p < 144) {
        pa = base + (size_t)(h * CHD + p / 12) * HW;
        pb = base + (size_t)(DIM + h * CHD + p % 12) * HW;
    } else if (p < 156) {
        pa = pb = base + (size_t)(h * CHD + (p - 144)) * HW;
    } else {
        pa = pb = base + (size_t)(DIM + h * CHD + (p - 156)) * HW;
    }
    float s = 0.f;
    for (int i = threadIdx.x; i < HW; i += 256) s += pa[i] * pb[i];
    float tot = block_reduce_sum(s, red);
    if (threadIdx.x == 0) G[(size_t)bh * 168 + p] = tot;
}

// ---------------- K8: normalize, modulate, softmax (12x12 per head) ----------------
__global__ void k_softmax(const float* __restrict__ G, const float* __restrict__ sgm,
                          const float* __restrict__ temperature, float* __restrict__ attn) {
    __shared__ float A[144];
    int bh = blockIdx.x;
    int b = bh / NH, h = bh % NH;
    int t = threadIdx.x;
    if (t < 144) {
        int c = t / 12, d = t % 12;
        float nq = fmaxf(sqrtf(G[(size_t)bh * 168 + 144 + c]), 1e-12f);
        float nk = fmaxf(sqrtf(G[(size_t)bh * 168 + 156 + d]), 1e-12f);
        A[t] = (G[(size_t)bh * 168 + t] / (nq * nk)) * temperature[h] *
               sgm[(size_t)b * 1152 + h * 144 + t];
    }
    __syncthreads();
    if (t < 12) {
        float mx = -1e30f;
        for (int d = 0; d < 12; d++) mx = fmaxf(mx, A[t * 12 + d]);
        float s = 0.f;
        for (int d = 0; d < 12; d++) {
            float e = expf(A[t * 12 + d] - mx);
            A[t * 12 + d] = e;
            s += e;
        }
        float inv = 1.0f / s;
        for (int d = 0; d < 12; d++) A[t * 12 + d] *= inv;
    }
    __syncthreads();
    if (t < 144) attn[(size_t)bh * 144 + t] = A[t];
}

// ---------------- K9: out = attn (12x12) @ v (12 x 25600) ----------------
__global__ void k_attnv(const float* __restrict__ attn, const float* __restrict__ qkv,
                        float* __restrict__ out_attn) {
    __shared__ float A[144];
    int bh = blockIdx.y;
    int b = bh / NH, h = bh % NH;
    int t = threadIdx.x;
    if (t < 144) A[t] = attn[(size_t)bh * 144 + t];
    __syncthreads();
    int n = blockIdx.x * 256 + t;
    const float* vb = qkv + (size_t)b * C3 * HW + (size_t)(2 * DIM + h * CHD) * HW + n;
    float vv[12];
#pragma unroll
    for (int d = 0; d < 12; d++) vv[d] = vb[(size_t)d * HW];
    float* ob = out_attn + (size_t)b * DIM * HW + (size_t)(h * CHD) * HW + n;
#pragma unroll
    for (int c = 0; c < 12; c++) {
        float acc = 0.f;
#pragma unroll
        for (int d = 0; d < 12; d++) acc += A[c * 12 + d] * vv[d];
        ob[(size_t)c * HW] = acc;
    }
}

extern "C" void kernel_launch(void* const* d_in, const int* in_sizes, int n_in,
                              void* d_out, int out_size, void* d_ws, size_t ws_size,
                              hipStream_t stream) {
    (void)in_sizes; (void)n_in; (void)out_size; (void)ws_size;
    const float* x           = (const float*)d_in[0];
    const float* spectral    = (const float*)d_in[1];
    const float* temperature = (const float*)d_in[2];
    const float* w_qkv       = (const float*)d_in[3];
    const float* w_dw        = (const float*)d_in[4];
    const float* w_proj      = (const float*)d_in[5];
    const float* w_fc        = (const float*)d_in[6];
    const float* b_fc        = (const float*)d_in[7];
    const float* w_dep       = (const float*)d_in[8];
    const float* b_dep       = (const float*)d_in[9];
    const float* w_se1       = (const float*)d_in[10];
    const float* b_se1       = (const float*)d_in[11];
    const float* w_se2       = (const float*)d_in[12];
    const float* b_se2       = (const float*)d_in[13];
    const float* w_saw1      = (const float*)d_in[14];
    const float* b_saw1      = (const float*)d_in[15];
    const float* w_saw2      = (const float*)d_in[16];
    const float* b_saw2      = (const float*)d_in[17];
    const float* w_sgp       = (const float*)d_in[18];
    const float* b_sgp       = (const float*)d_in[19];
    float* out = (float*)d_out;

    // workspace layout (floats)
    float* ws       = (float*)d_ws;
    float* chan_sum = ws;                       // 384
    float* sgm      = chan_sum + BD * DIM;      // 4608
    float* W_eff    = sgm + BD * 1152;          // 110592
    float* Gb       = W_eff + BD * C3 * DIM;    // 5376
    float* attn     = Gb + BD * NH * 168;       // 4608
    float* qkv_pre  = attn + BD * NH * 144;     // 29491200
    float* qkv      = qkv_pre + (size_t)BD * C3 * HW;   // 29491200
    float* f_conv   = qkv + (size_t)BD * C3 * HW;       // 11059200
    float* out_conv = f_conv + (size_t)BD * 108 * HW;   // 9830400
    float* out_attn = f_conv;   // alias: f_conv is dead after k_depconv

    // 1) per-channel sums (for cm)
    k_chansum<<<dim3(BD * DIM), dim3(256), 0, stream>>>(x, chan_sum);
    // 2) gates / folded weights / sgm
    k_prep<<<dim3(BD), dim3(256), 0, stream>>>(spectral, chan_sum, w_se1, b_se1, w_se2,
                                               b_se2, w_qkv, w_saw1, b_saw1, w_saw2,
                                               b_saw2, w_sgp, b_sgp, W_eff, sgm);
    // 3) qkv 1x1 conv as WMMA GEMM: [288x96] @ [96x25600] per batch
    k_wmma_gemm96<<<dim3((18 * (HW / 16)) / 8, BD), dim3(256), 0, stream>>>(
        W_eff, x, nullptr, qkv_pre,
        18, (long)(C3 * DIM), (long)DIM * HW, 0L, (long)C3 * HW);
    // 4) 3x3 depthwise
    k_dwconv<<<dim3(HW / 256, BD * C3), dim3(256), 0, stream>>>(qkv_pre, w_dw, qkv);
    // 5) fc branch
    k_fc<<<dim3(HW / 16, BD), dim3(192), 0, stream>>>(qkv, w_fc, b_fc, f_conv);
    // 6) grouped 3x3 conv
    k_depconv<<<dim3(HW / 256, BD * DIM), dim3(256), 0, stream>>>(f_conv, w_dep, b_dep,
                                                                  out_conv);
    // 7) gram matrices + norms
    k_gram<<<dim3(168, BD * NH), dim3(256), 0, stream>>>(qkv, Gb);
    // 8) softmax
    k_softmax<<<dim3(BD * NH), dim3(256), 0, stream>>>(Gb, sgm, temperature, attn);
    // 9) attn @ v
    k_attnv<<<dim3(HW / 256, BD * NH), dim3(256), 0, stream>>>(attn, qkv, out_attn);
    // 10) proj 1x1 as WMMA GEMM [96x96] @ [96x25600], fused + out_conv -> d_out
    k_wmma_gemm96<<<dim3((6 * (HW / 16)) / 8, BD), dim3(256), 0, stream>>>(
        w_proj, out_attn, out_conv, out,
        6, 0L, (long)DIM * HW, (long)DIM * HW, (long)DIM * HW);
}